// N1_LightGCN_39470749450553
// MI455X (gfx1250) — compile-verified
//
#include <hip/hip_runtime.h>
#include <hip/hip_bf16.h>

typedef __attribute__((ext_vector_type(2))) float v2f;
typedef __attribute__((ext_vector_type(8))) float v8f;

#define D_DIM 64

// ---------------------------------------------------------------- elementwise
__global__ void lg_copy_kernel(const float* __restrict__ in, float* __restrict__ out, long n) {
  long i = (long)blockIdx.x * blockDim.x + threadIdx.x;
  if (i < n) out[i] = in[i];
}

__global__ void lg_zero_kernel(float* __restrict__ p, long n) {
  long i = (long)blockIdx.x * blockDim.x + threadIdx.x;
  if (i < n) p[i] = 0.0f;
}

// e_cur = att[layer] * e_next ; acc += e_cur
__global__ void lg_scale_accum_kernel(const float* __restrict__ e_next,
                                      float* __restrict__ e_cur,
                                      float* __restrict__ acc,
                                      const float* __restrict__ att,
                                      int layer, long n) {
  long i = (long)blockIdx.x * blockDim.x + threadIdx.x;
  if (i < n) {
    float a = att[layer];
    float v = e_next[i] * a;
    e_cur[i] = v;
    acc[i] += v;
  }
}

__global__ void lg_scale_kernel(float* __restrict__ p, float s, long n) {
  long i = (long)blockIdx.x * blockDim.x + threadIdx.x;
  if (i < n) p[i] *= s;
}

// ---------------------------------------------------------------- COO SpMM scatter
// y[src[e], :] += val[e] * x[dst[e], :]   (D=64; 16 threads per edge, float4 each)
// Gathers are fully coalesced (16 consecutive lanes read one 256B row);
// scatter uses global_atomic_add_f32 resolved in L2 (table is L2-resident).
__global__ __launch_bounds__(256) void lg_spmm_scatter_kernel(
    const int* __restrict__ src, const int* __restrict__ dst,
    const float* __restrict__ val, const float* __restrict__ x,
    float* __restrict__ y, long nE) {
  long t = (long)blockIdx.x * blockDim.x + threadIdx.x;
  long e = t >> 4;
  if (e >= nE) return;
  int seg = (int)(t & 15);
  int s = src[e];
  int d = dst[e];
  float v = val[e];
  const float4* xr = (const float4*)(x + (long)d * D_DIM) + seg;
  float4 q = *xr;
  float* yr = y + (long)s * D_DIM + seg * 4;
  atomicAdd(yr + 0, v * q.x);
  atomicAdd(yr + 1, v * q.y);
  atomicAdd(yr + 2, v * q.z);
  atomicAdd(yr + 3, v * q.w);
}

// ---------------------------------------------------------------- batched dot via WMMA
// One wave32 per 16 (user,item) pairs. C += A(16x4,f32) x B(4x16,f32) chained
// over 16 K-chunks (K=64 total) with V_WMMA_F32_16X16X4_F32; gamma_r = diag(C).
//
// A layout (ISA 7.12.2, 32-bit A 16x4): lane m(0-15): VGPR0=K0,VGPR1=K1;
//   lane m+16: VGPR0=K2,VGPR1=K3.   => per lane: m=lane&15, kb=(lane>>4)*2.
// B layout (row striped across lanes): lane n(0-15): VGPR0=B[0][n],VGPR1=B[1][n];
//   lane n+16: B[2][n],B[3][n]. With B[k][n]=item_n[k] this is the same per-lane
//   indexing as A but on the item rows.
// C/D layout: lanes 0-15 N=lane, VGPR v = M=v; lanes 16-31 N=lane-16, M=v+8.
//   diag r<8  -> lane r,    element c[r]
//   diag r>=8 -> lane r+16, element c[r-8]
__global__ __launch_bounds__(32) void lg_gamma_wmma_kernel(
    const int* __restrict__ users, const int* __restrict__ items,
    const float* __restrict__ uemb, const float* __restrict__ iemb,
    float* __restrict__ out, int B) {
  int base = blockIdx.x * 16;
  int lane = threadIdx.x;
  int m = lane & 15;
  int kb = (lane >> 4) << 1;  // 0 or 2
  int p = base + m;
  if (p >= B) p = B - 1;      // keep EXEC all-ones for WMMA
  const float* up = uemb + (long)users[p] * D_DIM;
  const float* ip = iemb + (long)items[p] * D_DIM;

  v8f c = {};
#pragma unroll
  for (int t = 0; t < 16; ++t) {
    int k = t * 4 + kb;
    v2f a;
    a.x = up[k];
    a.y = up[k + 1];
    v2f b;
    b.x = ip[k];
    b.y = ip[k + 1];
    // 8 args: (neg_a, A, neg_b, B, c_mod, C, reuse_a, reuse_b)
    c = __builtin_amdgcn_wmma_f32_16x16x4_f32(false, a, false, b, (short)0, c,
                                              false, false);
  }

  // extract diagonal
  bool writer = false;
  int r = 0, idx = 0;
  if (lane < 8) {
    writer = true; r = lane; idx = lane;
  } else if (lane >= 24) {
    writer = true; r = lane - 16; idx = lane - 24;
  }
  float g = c[0];
#pragma unroll
  for (int j = 1; j < 8; ++j) g = (idx == j) ? c[j] : g;
  if (writer && (base + r) < B) out[base + r] = g;
}

// ---------------------------------------------------------------- launch
extern "C" void kernel_launch(void* const* d_in, const int* in_sizes, int n_in,
                              void* d_out, int out_size, void* d_ws, size_t ws_size,
                              hipStream_t stream) {
  const int*   users    = (const int*)d_in[0];
  const int*   items    = (const int*)d_in[1];
  const int*   ii_src   = (const int*)d_in[2];
  const int*   ii_dst   = (const int*)d_in[3];
  const float* ii_val   = (const float*)d_in[4];
  const int*   ui_src   = (const int*)d_in[5];
  const int*   ui_dst   = (const int*)d_in[6];
  const float* ui_val   = (const float*)d_in[7];
  const float* item_emb = (const float*)d_in[8];
  const float* att      = (const float*)d_in[9];

  const long E_ii = in_sizes[2];
  const long E_ui = in_sizes[5];
  const long MD   = in_sizes[8];      // M * 64
  const int  L    = in_sizes[9];
  const int  B    = in_sizes[0];
  const long U_ROWS = 50000;          // n_users: fixed by workload, not derivable from sizes
  const long UD = U_ROWS * D_DIM;

  // workspace partition (floats): e_cur | e_next | acc | user_emb  (~90 MB)
  float* e_cur  = (float*)d_ws;
  float* e_next = e_cur + MD;
  float* acc    = e_next + MD;
  float* uemb   = acc + MD;
  (void)ws_size; (void)n_in; (void)out_size;

  auto cdiv = [](long a, long b) { return (unsigned)((a + b - 1) / b); };

  // acc = item_emb (layer-0 embedding)
  lg_copy_kernel<<<cdiv(MD, 256), 256, 0, stream>>>(item_emb, acc, MD);

  const float* ein = item_emb;
  for (int l = 0; l < L; ++l) {
    lg_zero_kernel<<<cdiv(MD, 256), 256, 0, stream>>>(e_next, MD);
    lg_spmm_scatter_kernel<<<cdiv(E_ii * 16, 256), 256, 0, stream>>>(
        ii_src, ii_dst, ii_val, ein, e_next, E_ii);
    lg_scale_accum_kernel<<<cdiv(MD, 256), 256, 0, stream>>>(
        e_next, e_cur, acc, att, l, MD);
    ein = e_cur;
  }

  // all_items = mean over (L+1) embeddings
  lg_scale_kernel<<<cdiv(MD, 256), 256, 0, stream>>>(acc, 1.0f / (float)(L + 1), MD);

  // all_users = spmm(ui) @ all_items
  lg_zero_kernel<<<cdiv(UD, 256), 256, 0, stream>>>(uemb, UD);
  lg_spmm_scatter_kernel<<<cdiv(E_ui * 16, 256), 256, 0, stream>>>(
      ui_src, ui_dst, ui_val, acc, uemb, E_ui);

  // gamma[b] = dot(all_users[users[b]], all_items[items[b]])  -- WMMA f32
  lg_gamma_wmma_kernel<<<cdiv(B, 16), 32, 0, stream>>>(
      users, items, uemb, acc, (float*)d_out, B);
}